// WaveletBlock_81681688036025
// MI455X (gfx1250) — compile-verified
//
#include <hip/hip_runtime.h>
#include <hip/hip_bf16.h>

// ---------------------------------------------------------------------------
// WaveletBlock for MI455X (gfx1250).
// Memory-bound fusion: one workgroup owns one (b,c) 256x256 plane and runs the
// whole 3-level wavelet pipeline in 320KB LDS. TDM (tensor_load_to_lds) stages
// x row-strips into LDS; v_wmma_f32_16x16x4_f32 does the block stats reduction.
// All halo taps use masked (wrapped) indices so every ds_load is unconditional
// (select-zero on the value instead of exec-mask branches).
// ---------------------------------------------------------------------------

typedef float v2f __attribute__((ext_vector_type(2)));
typedef float v8f __attribute__((ext_vector_type(8)));
typedef unsigned int u32x4 __attribute__((ext_vector_type(4)));
typedef int i32x4 __attribute__((ext_vector_type(4)));
typedef int i32x8 __attribute__((ext_vector_type(8)));

#if defined(__has_builtin)
#if __has_builtin(__builtin_amdgcn_tensor_load_to_lds) && __has_builtin(__builtin_amdgcn_s_wait_tensorcnt)
#define HAVE_TDM 1
#endif
#if __has_builtin(__builtin_amdgcn_wmma_f32_16x16x4_f32)
#define HAVE_WMMA_F32X4 1
#endif
#endif

#define B_ 16
#define C_ 64
#define H_ 256
#define W_ 256
#define PLANE (H_ * W_) // 65536

// LDS layout in floats (dynamic shared). Peak = phase C: A1 + XS + SBS.
#define OFF_A1 0       // 16384  LL0 -> SB1 (interleaved) -> nxt1
#define OFF_T1 16384   // 16384  t1 (interleaved, level-1 conv out)
#define OFF_A2 32768   // 4096   LL1 -> SB2 (interleaved)
#define OFF_T2 36864   // 4096   t2
#define OFF_N2 40960   // 4096   nxt2
#define OFF_XS 16384   // 17408  phase-C x strip (68 rows x 256) -- reuses T1
#define OFF_SBS 33792  // 17408  phase-C SB0 strip (34 sb-rows interleaved)
#define OFF_WGT 51200  // 131    staged per-channel weights
#define OFF_RED 51332  // 16     per-wave partial sums
#define LDS_FLOATS 51352

// ---------------------------------------------------------------------------
// TDM: build a 2D tile descriptor (D#) per CDNA5 ISA ch.8 and issue
// tensor_load_to_lds (6-arg clang-23 form: g0, g1, g2, g3, spare, cpol).
// rows x 256-float rows, contiguous in both global & LDS.
// ---------------------------------------------------------------------------
__device__ __forceinline__ void tdm_load_rows(const float* gsrc,
                                              unsigned lds_byte, int rows) {
#if defined(HAVE_TDM)
  unsigned long long ga = (unsigned long long)(const void*)gsrc;
  u32x4 g0;
  g0[0] = 1u;                                   // count=1, user mode
  g0[1] = lds_byte;                             // lds_addr [63:32]
  g0[2] = (unsigned)(ga & 0xFFFFFFFFull);       // global_addr lo
  g0[3] = (unsigned)((ga >> 32) & 0x1FFFFFFull) // global_addr[56:32]
          | (2u << 30);                         // type=2 (image)
  i32x8 g1;
  g1[0] = (2 << 16);                      // data_size=4B; no multicast/pad
  g1[1] = (int)((unsigned)W_ << 16);      // tensor_dim0[15:0] @bit48
  g1[2] = (int)(((unsigned)rows) << 16);  // tensor_dim0 hi=0 | tensor_dim1 lo
  g1[3] = (int)((unsigned)W_ << 16);      // tensor_dim1 hi=0 | tile_dim0=256
  g1[4] = rows & 0xFFFF;                  // tile_dim1=rows, tile_dim2=0
  g1[5] = W_;                             // tensor_dim0_stride lo32 = 256
  g1[6] = 0;                              // stride hi | dim1_stride lo
  g1[7] = 0;
  i32x4 g2; g2[0] = 1; g2[1] = 0; g2[2] = 0; g2[3] = 0; // tensor_dim2=1
  i32x4 g3; g3[0] = 0; g3[1] = 0; g3[2] = 0; g3[3] = 0;
  i32x8 g4; g4[0] = 0; g4[1] = 0; g4[2] = 0; g4[3] = 0;
  g4[4] = 0; g4[5] = 0; g4[6] = 0; g4[7] = 0;
  __builtin_amdgcn_tensor_load_to_lds(g0, g1, g2, g3, g4, 0);
#else
  (void)gsrc; (void)lds_byte; (void)rows;
#endif
}

// ---------------------------------------------------------------------------
// K0: zero channel accumulators (sum, sumsq per channel) in workspace.
// ---------------------------------------------------------------------------
__global__ void k_zero(float* accum) {
  if (threadIdx.x < 2 * C_) accum[threadIdx.x] = 0.0f;
}

// ---------------------------------------------------------------------------
// K1: per-plane wavelet pipeline. grid = B*C blocks of 256 threads.
// ---------------------------------------------------------------------------
__global__ void k_wavelet(const float* __restrict__ x,
                          const float* __restrict__ base_w,
                          const float* __restrict__ base_b,
                          const float* __restrict__ base_scale,
                          const float* __restrict__ ww0,
                          const float* __restrict__ ww1,
                          const float* __restrict__ ww2,
                          const float* __restrict__ ws0,
                          const float* __restrict__ ws1,
                          const float* __restrict__ ws2,
                          float* __restrict__ accum,
                          float* __restrict__ yout) {
  extern __shared__ float lds[];
  const int plane = blockIdx.x;
  const int c = plane & (C_ - 1);
  const int tid = threadIdx.x;
  const float* xp = x + (size_t)plane * PLANE;
  float* yp = yout + (size_t)plane * PLANE;

  // ---- stage per-channel weights into LDS -------------------------------
  // [0..8] base_w, [9] base_b, [10] base_scale,
  // [11..46] ww0 (4 bands x 9), [47..50] ws0,
  // [51..86] ww1, [87..90] ws1, [91..126] ww2, [127..130] ws2.
  for (int t = tid; t < 131; t += 256) {
    float v;
    if (t < 9)        v = base_w[c * 9 + t];
    else if (t == 9)  v = base_b[c];
    else if (t == 10) v = base_scale[c];
    else if (t < 47)  v = ww0[c * 36 + (t - 11)];
    else if (t < 51)  v = ws0[c * 4 + (t - 47)];
    else if (t < 87)  v = ww1[c * 36 + (t - 51)];
    else if (t < 91)  v = ws1[c * 4 + (t - 87)];
    else if (t < 127) v = ww2[c * 36 + (t - 91)];
    else              v = ws2[c * 4 + (t - 127)];
    lds[OFF_WGT + t] = v;
  }

  // ---- Phase A: LL0 (raw level-0 low-low, 128x128) from global x --------
  for (int m = 0; m < 64; ++m) {
    int e = tid + m * 256;           // 16384 elements
    int i = e >> 7, j = e & 127;
    const float* r0 = xp + (2 * i) * W_ + 2 * j;
    float2 a = *(const float2*)r0;
    float2 b = *(const float2*)(r0 + W_);
    lds[OFF_A1 + i * 128 + j] = 0.5f * (a.x + a.y + b.x + b.y);
  }
  __syncthreads();

  // ---- B1: in-place DWT of LL0 -> SB1 interleaved (4 x 64x64) -----------
  for (int m = 0; m < 16; ++m) {
    int e = tid + m * 256;           // 4096 blocks
    int i = e >> 6, j = e & 63;
    float* p00 = &lds[OFF_A1 + (2 * i) * 128 + 2 * j];
    float a = p00[0], b = p00[1], cc = p00[128], d = p00[129];
    p00[0]   = 0.5f * (a + b + cc + d);   // band0 (LL)
    p00[1]   = 0.5f * (a + b - cc - d);   // band1
    p00[128] = 0.5f * (a - b + cc - d);   // band2
    p00[129] = 0.5f * (a - b - cc + d);   // band3
  }
  __syncthreads();

  // ---- B2: level-1 depthwise conv (SB1 -> T1) + B3: extract raw LL1 -----
  for (int m = 0; m < 64; ++m) {
    int e = tid + m * 256;           // 4 bands x 64x64
    int k = e >> 12, r = e & 4095, i = r >> 6, j = r & 63;
    float acc = 0.0f;
#pragma unroll
    for (int du = -1; du <= 1; ++du) {
      int ii = i + du;
#pragma unroll
      for (int dv = -1; dv <= 1; ++dv) {
        int jj = j + dv;
        bool ok = ((unsigned)ii < 64u) && ((unsigned)jj < 64u);
        float xv = lds[OFF_A1 + (2 * (ii & 63) + (k >> 1)) * 128 +
                       2 * (jj & 63) + (k & 1)];
        acc += (ok ? xv : 0.0f) *
               lds[OFF_WGT + 51 + k * 9 + (du + 1) * 3 + (dv + 1)];
      }
    }
    lds[OFF_T1 + (2 * i + (k >> 1)) * 128 + 2 * j + (k & 1)] =
        acc * lds[OFF_WGT + 87 + k];
  }
  for (int m = 0; m < 16; ++m) {     // raw LL1 -> A2
    int e = tid + m * 256;
    int i = e >> 6, j = e & 63;
    lds[OFF_A2 + i * 64 + j] = lds[OFF_A1 + (2 * i) * 128 + 2 * j];
  }
  __syncthreads();

  // ---- B4: in-place DWT of LL1 -> SB2 interleaved (4 x 32x32) -----------
  for (int m = 0; m < 4; ++m) {
    int e = tid + m * 256;           // 1024 blocks
    int i = e >> 5, j = e & 31;
    float* p00 = &lds[OFF_A2 + (2 * i) * 64 + 2 * j];
    float a = p00[0], b = p00[1], cc = p00[64], d = p00[65];
    p00[0]  = 0.5f * (a + b + cc + d);
    p00[1]  = 0.5f * (a + b - cc - d);
    p00[64] = 0.5f * (a - b + cc - d);
    p00[65] = 0.5f * (a - b - cc + d);
  }
  __syncthreads();

  // ---- B5: level-2 depthwise conv (SB2 -> T2) ---------------------------
  for (int m = 0; m < 16; ++m) {
    int e = tid + m * 256;           // 4 bands x 32x32
    int k = e >> 10, r = e & 1023, i = r >> 5, j = r & 31;
    float acc = 0.0f;
#pragma unroll
    for (int du = -1; du <= 1; ++du) {
      int ii = i + du;
#pragma unroll
      for (int dv = -1; dv <= 1; ++dv) {
        int jj = j + dv;
        bool ok = ((unsigned)ii < 32u) && ((unsigned)jj < 32u);
        float xv = lds[OFF_A2 + (2 * (ii & 31) + (k >> 1)) * 64 +
                       2 * (jj & 31) + (k & 1)];
        acc += (ok ? xv : 0.0f) *
               lds[OFF_WGT + 91 + k * 9 + (du + 1) * 3 + (dv + 1)];
      }
    }
    lds[OFF_T2 + (2 * i + (k >> 1)) * 64 + 2 * j + (k & 1)] =
        acc * lds[OFF_WGT + 127 + k];
  }
  __syncthreads();

  // ---- B6: idwt level-2 -> nxt2 (64x64) ---------------------------------
  for (int m = 0; m < 4; ++m) {
    int e = tid + m * 256;           // 1024 blocks
    int i = e >> 5, j = e & 31;
    const float* p00 = &lds[OFF_T2 + (2 * i) * 64 + 2 * j];
    float t0 = p00[0], t1 = p00[1], t2 = p00[64], t3 = p00[65];
    float* n = &lds[OFF_N2 + (2 * i) * 64 + 2 * j];
    n[0]  = 0.5f * (t0 + t1 + t2 + t3);
    n[1]  = 0.5f * (t0 + t1 - t2 - t3);
    n[64] = 0.5f * (t0 - t1 + t2 - t3);
    n[65] = 0.5f * (t0 - t1 - t2 + t3);
  }
  __syncthreads();

  // ---- B7a: T1 band0 += nxt2 --------------------------------------------
  for (int m = 0; m < 16; ++m) {
    int e = tid + m * 256;
    int i = e >> 6, j = e & 63;
    lds[OFF_T1 + (2 * i) * 128 + 2 * j] += lds[OFF_N2 + i * 64 + j];
  }
  __syncthreads();

  // ---- B7b: idwt level-1 -> nxt1 (128x128) into A1 ----------------------
  for (int m = 0; m < 16; ++m) {
    int e = tid + m * 256;
    int i = e >> 6, j = e & 63;
    const float* p00 = &lds[OFF_T1 + (2 * i) * 128 + 2 * j];
    float t0 = p00[0], t1 = p00[1], t2 = p00[128], t3 = p00[129];
    float* n = &lds[OFF_A1 + (2 * i) * 128 + 2 * j];
    n[0]   = 0.5f * (t0 + t1 + t2 + t3);
    n[1]   = 0.5f * (t0 + t1 - t2 - t3);
    n[128] = 0.5f * (t0 - t1 + t2 - t3);
    n[129] = 0.5f * (t0 - t1 - t2 + t3);
  }
  __syncthreads();

  // ---- Phase C: 4 strips of 32 subband rows -----------------------------
  float sAcc = 0.0f, qAcc = 0.0f;
  for (int s = 0; s < 4; ++s) {
    const int s0 = s * 32;
    const int xr0 = 2 * s0 - 2;      // first x row in strip buffer (68 rows)
    const int v0 = xr0 < 0 ? 0 : xr0;
    const int v1 = (xr0 + 68 > H_) ? H_ : (xr0 + 68);
#if defined(HAVE_TDM)
    for (int e = tid; e < 68 * W_; e += 256) {  // zero out-of-plane rows
      int gr = xr0 + (e >> 8);
      if (gr < 0 || gr >= H_) lds[OFF_XS + e] = 0.0f;
    }
    if (tid == 0) {
      unsigned ldsb =
          (unsigned)(unsigned long long)(const void*)&lds[OFF_XS + (v0 - xr0) * W_];
      tdm_load_rows(xp + v0 * W_, ldsb, v1 - v0);
      __builtin_amdgcn_s_wait_tensorcnt(0);
    }
#else
    for (int e = tid; e < 68 * W_; e += 256) {
      int gr = xr0 + (e >> 8);
      lds[OFF_XS + e] = (gr >= 0 && gr < H_) ? xp[gr * W_ + (e & 255)] : 0.0f;
    }
#endif
    __syncthreads();

    // C3: SB0 strip rows [s0-1 .. s0+33) interleaved
    for (int m = 0; m < 17; ++m) {
      int e = tid + m * 256;         // 34*128 = 4352
      int sbl = e >> 7, j = e & 127;
      const float* X0 = &lds[OFF_XS + (2 * sbl) * W_ + 2 * j];
      float a = X0[0], b = X0[1], cc = X0[W_], d = X0[W_ + 1];
      float* S = &lds[OFF_SBS + (2 * sbl) * W_ + 2 * j];
      S[0]      = 0.5f * (a + b + cc + d);
      S[1]      = 0.5f * (a + b - cc - d);
      S[W_]     = 0.5f * (a - b + cc - d);
      S[W_ + 1] = 0.5f * (a - b - cc + d);
    }
    __syncthreads();

    // C4: level-0 conv + nxt1 + final idwt + base conv -> y
    for (int m = 0; m < 16; ++m) {
      int e = tid + m * 256;         // 32*128 = 4096 quads
      int il = e >> 7, j = e & 127;
      int i = s0 + il;
      float t[4];
#pragma unroll
      for (int k = 0; k < 4; ++k) {
        float acc = 0.0f;
#pragma unroll
        for (int du = -1; du <= 1; ++du) {
          const float* S =
              &lds[OFF_SBS + (2 * (il + 1 + du) + (k >> 1)) * W_ + (k & 1)];
#pragma unroll
          for (int dv = -1; dv <= 1; ++dv) {
            int jj = j + dv;
            float xv = S[2 * (jj & 127)];          // unconditional ds_load
            acc += (((unsigned)jj < 128u) ? xv : 0.0f) *
                   lds[OFF_WGT + 11 + k * 9 + (du + 1) * 3 + (dv + 1)];
          }
        }
        t[k] = acc * lds[OFF_WGT + 47 + k];
      }
      t[0] += lds[OFF_A1 + i * 128 + j];

      float yq[4];
#pragma unroll
      for (int p = 0; p < 2; ++p) {
#pragma unroll
        for (int q = 0; q < 2; ++q) {
          float acc = 0.0f;
#pragma unroll
          for (int du = -1; du <= 1; ++du) {
            const float* X = &lds[OFF_XS + (2 * il + 2 + p + du) * W_];
#pragma unroll
            for (int dv = -1; dv <= 1; ++dv) {
              int ccol = 2 * j + q + dv;
              float xv = X[ccol & 255];            // unconditional ds_load
              acc += (((unsigned)ccol < 256u) ? xv : 0.0f) *
                     lds[OFF_WGT + (du + 1) * 3 + (dv + 1)];
            }
          }
          float base = (acc + lds[OFF_WGT + 9]) * lds[OFF_WGT + 10];
          float w;
          if (p == 0 && q == 0)      w = 0.5f * (t[0] + t[1] + t[2] + t[3]);
          else if (p == 0)           w = 0.5f * (t[0] + t[1] - t[2] - t[3]);
          else if (q == 0)           w = 0.5f * (t[0] - t[1] + t[2] - t[3]);
          else                       w = 0.5f * (t[0] - t[1] - t[2] + t[3]);
          float yv = base + w;
          yq[p * 2 + q] = yv;
          sAcc += yv;
          qAcc += yv * yv;
        }
      }
      float* orow = yp + (2 * i) * W_ + 2 * j;
      *(float2*)orow = make_float2(yq[0], yq[1]);
      *(float2*)(orow + W_) = make_float2(yq[2], yq[3]);
    }
    __syncthreads();
  }

  // ---- stats reduction: WMMA f32 16x16x4 matrix-engine reduction --------
  float s = sAcc, q = qAcc;
#if defined(HAVE_WMMA_F32X4)
  {
    // A[M,K]: lane L<16 -> (M=L, K=0/1), lane L>=16 -> (M=L-16, K=2/3).
    // Put sum in K in {0,2}, sumsq in K in {1,3}; selector B picks each set:
    // D0[m,n] = sum_m + sum_{m+16}, D1[m,n] = sq_m + sq_{m+16}.
    v2f a;  a.x = s;    a.y = q;
    v2f b0; b0.x = 1.f; b0.y = 0.f;
    v2f b1; b1.x = 0.f; b1.y = 1.f;
    v8f cz = {0.f, 0.f, 0.f, 0.f, 0.f, 0.f, 0.f, 0.f};
    v8f d0 = __builtin_amdgcn_wmma_f32_16x16x4_f32(false, a, false, b0,
                                                   (short)0, cz, false, false);
    v8f d1 = __builtin_amdgcn_wmma_f32_16x16x4_f32(false, a, false, b1,
                                                   (short)0, cz, false, false);
    s = d0[0] + d0[1] + d0[2] + d0[3] + d0[4] + d0[5] + d0[6] + d0[7];
    q = d1[0] + d1[1] + d1[2] + d1[3] + d1[4] + d1[5] + d1[6] + d1[7];
    s += __shfl_xor(s, 16, 32);  // rows 0-7 half + rows 8-15 half
    q += __shfl_xor(q, 16, 32);
  }
#else
  for (int o = 16; o > 0; o >>= 1) {
    s += __shfl_xor(s, o, 32);
    q += __shfl_xor(q, o, 32);
  }
#endif
  if ((tid & 31) == 0) {
    lds[OFF_RED + (tid >> 5)] = s;
    lds[OFF_RED + 8 + (tid >> 5)] = q;
  }
  __syncthreads();
  if (tid == 0) {
    float ts = 0.0f, tq = 0.0f;
    for (int w = 0; w < 8; ++w) {
      ts += lds[OFF_RED + w];
      tq += lds[OFF_RED + 8 + w];
    }
    atomicAdd(&accum[2 * c], ts);
    atomicAdd(&accum[2 * c + 1], tq);
  }
}

// ---------------------------------------------------------------------------
// K2: per-channel normalize + gamma/beta + relu, in place on d_out (float4).
// ---------------------------------------------------------------------------
__global__ void k_norm(float* __restrict__ y, const float* __restrict__ accum,
                       const float* __restrict__ gamma,
                       const float* __restrict__ beta) {
  long long t = (long long)blockIdx.x * 256 + threadIdx.x;
  long long e4 = t * 4;
  int c = (int)((e4 >> 16) & (C_ - 1));
  const float invN = 1.0f / (float)((long long)B_ * H_ * W_);
  float mean = accum[2 * c] * invN;
  float var = accum[2 * c + 1] * invN - mean * mean;
  var = var < 0.0f ? 0.0f : var;
  float inv = rsqrtf(var + 1e-5f);
  float g = gamma[c] * inv;
  float bb = beta[c] - mean * g;
  float4 v = *(float4*)(y + e4);
  v.x = fmaxf(v.x * g + bb, 0.0f);
  v.y = fmaxf(v.y * g + bb, 0.0f);
  v.z = fmaxf(v.z * g + bb, 0.0f);
  v.w = fmaxf(v.w * g + bb, 0.0f);
  *(float4*)(y + e4) = v;
}

// ---------------------------------------------------------------------------
extern "C" void kernel_launch(void* const* d_in, const int* in_sizes, int n_in,
                              void* d_out, int out_size, void* d_ws,
                              size_t ws_size, hipStream_t stream) {
  const float* x          = (const float*)d_in[0];
  const float* base_w     = (const float*)d_in[1];
  const float* base_b     = (const float*)d_in[2];
  const float* base_scale = (const float*)d_in[3];
  const float* ww0        = (const float*)d_in[4];
  const float* ww1        = (const float*)d_in[5];
  const float* ww2        = (const float*)d_in[6];
  const float* ws0        = (const float*)d_in[7];
  const float* ws1        = (const float*)d_in[8];
  const float* ws2        = (const float*)d_in[9];
  const float* gamma      = (const float*)d_in[10];
  const float* beta       = (const float*)d_in[11];
  float* out = (float*)d_out;
  float* accum = (float*)d_ws;

  k_zero<<<1, 128, 0, stream>>>(accum);
  k_wavelet<<<dim3(B_ * C_), dim3(256), LDS_FLOATS * sizeof(float), stream>>>(
      x, base_w, base_b, base_scale, ww0, ww1, ww2, ws0, ws1, ws2, accum, out);
  long long total = (long long)B_ * C_ * PLANE;
  k_norm<<<dim3((unsigned)(total / (256 * 4))), dim3(256), 0, stream>>>(
      out, accum, gamma, beta);
}